// exponential_smoothing_69793218560068
// MI455X (gfx1250) — compile-verified
//
#include <hip/hip_runtime.h>

// EMA scan (alpha = 0.5) over (B=32, T=4096, C=512) fp32, HBM-bound (~23 us at 23.3 TB/s).
// Strategy: 16-step scan blocks as triangular 16x16 matmuls on V_WMMA_F32_16X16X4_F32,
// carry injected as rank-1 update, T split into 8 segments with 64-step warm-up
// (carry attenuation 2^-64, exact powers of two coefficients).

typedef __attribute__((ext_vector_type(2))) float v2f;
typedef __attribute__((ext_vector_type(8))) float v8f;

namespace {
constexpr int BATCH = 32;
constexpr int T     = 4096;
constexpr int C     = 512;
constexpr int TSEG  = 512;        // timesteps per independent segment
constexpr int NSEG  = T / TSEG;   // 8
constexpr int WARM  = 64;         // warm-up steps: carry error attenuated by 2^-64
constexpr int CH    = 16;         // WMMA chunk (timesteps per matmul block)
constexpr int WAVES_PER_BLOCK = 8;
constexpr int TOTAL_WAVES = BATCH * (C / 16) * NSEG;  // 32*32*8 = 8192
}

__global__ __launch_bounds__(256) void ema_scan_wmma_kernel(
    const float* __restrict__ x, float* __restrict__ out) {
  const int lane  = threadIdx.x & 31;
  const int w     = blockIdx.x * WAVES_PER_BLOCK + (threadIdx.x >> 5);
  const int b     = w >> 8;          // batch       (32)
  const int cs    = (w >> 3) & 31;   // channel strip (32 strips of 16)
  const int seg   = w & 7;           // time segment (8)

  const int khalf = lane >> 4;       // lane half selects K offset / row+8
  const int i     = lane & 15;       // M (A rows) == N (B/C/D cols) index
  const int c0    = cs * 16;

  // ---- A matrix: lower-triangular EMA coefficients, 4 K-slices of 16x4 ----
  // A slice s, vgpr v, lane L holds M=i, K = 4s + 2*khalf + v.
  v2f a[4];
#pragma unroll
  for (int s = 0; s < 4; ++s) {
#pragma unroll
    for (int v = 0; v < 2; ++v) {
      const int j = 4 * s + 2 * khalf + v;
      a[s][v] = (j <= i) ? exp2f(-(float)(i - j + 1)) : 0.0f;  // 2^-(i-j+1)
    }
  }
  // Carry-injection coefficients per C/D row: d[r] = 2^-(M+1), M = r + 8*khalf
  float d[8];
  {
    float dr = khalf ? 0x1.0p-9f : 0x1.0p-1f;
#pragma unroll
    for (int r = 0; r < 8; ++r) { d[r] = dr; dr *= 0.5f; }
  }

  const int tstart = seg * TSEG;
  const int nwarm  = (seg == 0) ? 0 : (WARM / CH);
  const int nchunk = nwarm + TSEG / CH;
  const int tbase  = tstart - nwarm * CH;

  const float* xw = x   + (size_t)b * T * C + c0;
  float*       ow = out + (size_t)b * T * C + c0;

  // Carry per lane = y[t0-1] for channel c0+i, replicated in both lane halves.
  // Segment 0 uses carry = x[b,0,c]: row 0 then yields 2^-1*x0 + 2^-1*x0 = x0 exactly.
  float vcarry = (seg == 0) ? xw[i] : 0.0f;

  // B tile: slice s, vgpr v, lane L holds X[t0 + 4s + 2*khalf + v][c0 + i]
  auto loadB = [&](int t0, v2f bm[4]) {
    const float* p = xw + (size_t)(t0 + 2 * khalf) * C + i;
#pragma unroll
    for (int s = 0; s < 4; ++s) {
      bm[s][0] = p[(size_t)(4 * s) * C];
      bm[s][1] = p[(size_t)(4 * s + 1) * C];
    }
  };

  v2f bcur[4], bnext[4];
  loadB(tbase, bcur);

  for (int k = 0; k < nchunk; ++k) {
    const int t0 = tbase + k * CH;
    if (k + 1 < nchunk) loadB(t0 + CH, bnext);   // double-buffer next tile

    // Y_partial = M * X  (16x16x16 via four K=4 WMMAs, fp32 accumulate)
    v8f acc = {};
#pragma unroll
    for (int s = 0; s < 4; ++s) {
      acc = __builtin_amdgcn_wmma_f32_16x16x4_f32(
          /*neg_a=*/false, a[s], /*neg_b=*/false, bcur[s],
          /*c_mod=*/(short)0, acc, /*reuse_a=*/false, /*reuse_b=*/false);
    }

    // Y = Y_partial + d * carry (rank-1 carry injection)
    v8f y;
#pragma unroll
    for (int r = 0; r < 8; ++r) y[r] = fmaf(d[r], vcarry, acc[r]);

    // New carry = row M=15 -> VGPR 7, lanes 16..31 hold N=0..15
    vcarry = __shfl(y[7], 16 + i, 32);

    if (k >= nwarm) {
      float* po = ow + (size_t)(t0 + khalf * 8) * C + i;
#pragma unroll
      for (int r = 0; r < 8; ++r) po[(size_t)r * C] = y[r];  // 64B segments/half-wave
    }

#pragma unroll
    for (int s = 0; s < 4; ++s) bcur[s] = bnext[s];
  }
}

extern "C" void kernel_launch(void* const* d_in, const int* in_sizes, int n_in,
                              void* d_out, int out_size, void* d_ws, size_t ws_size,
                              hipStream_t stream) {
  (void)in_sizes; (void)n_in; (void)d_ws; (void)ws_size; (void)out_size;
  const float* x = (const float*)d_in[0];
  float* out = (float*)d_out;
  const int blocks = TOTAL_WAVES / WAVES_PER_BLOCK;  // 1024 blocks of 256 threads
  ema_scan_wmma_kernel<<<blocks, WAVES_PER_BLOCK * 32, 0, stream>>>(x, out);
}